// SelectiveSSM_64553358459349
// MI455X (gfx1250) — compile-verified
//
#include <hip/hip_runtime.h>
#include <hip/hip_bf16.h>
#include <math.h>

// SelectiveSSM (Mamba-style) for MI455X gfx1250, wave32.
// B=2, L=2048, D=2048, N=16, R=128.

typedef __attribute__((ext_vector_type(2))) float v2f;
typedef __attribute__((ext_vector_type(8))) float v8f;

#define BB   2
#define LL   2048
#define DD   2048
#define NN   16
#define RR   128
#define PP   (RR + 2 * NN)   // 160 = x_proj output width
#define MM   (BB * LL)       // 4096 rows

// ---------------------------------------------------------------------------
// Kernel 1: xp[M, 160] = x[M, 2048] @ x_proj_w[160, 2048]^T  (fp32 WMMA)
// One wave32 per 16x16 output tile; K-loop in steps of 4 with
// V_WMMA_F32_16X16X4_F32.
// ---------------------------------------------------------------------------
__global__ __launch_bounds__(32) void ssm_xproj_wmma(
    const float* __restrict__ X,   // [MM, DD]
    const float* __restrict__ W,   // [PP, DD]
    float* __restrict__ XP) {      // [MM, PP]
  const int mt   = blockIdx.x;     // 0..255  (M tiles)
  const int pt   = blockIdx.y;     // 0..9    (P tiles)
  const int lane = threadIdx.x;    // 0..31
  const int half = lane >> 4;      // lane-half (selects K pair / M+8)
  const int l16  = lane & 15;

  const float* xrow = X + (size_t)(mt * 16 + l16) * DD;  // A row = M
  const float* wrow = W + (size_t)(pt * 16 + l16) * DD;  // B col = N (w row)

  v8f acc = {};
  for (int k = 0; k < DD; k += 4) {
    // A 16x4 fp32: VGPR j holds K = 2*half + j (contiguous pair per lane)
    v2f a = *(const v2f*)(xrow + k + half * 2);
    // B 4x16 fp32: VGPR j holds K = 2*j + half, N = l16
    v2f b;
    b.x = wrow[k + half];
    b.y = wrow[k + 2 + half];
    acc = __builtin_amdgcn_wmma_f32_16x16x4_f32(
        false, a, false, b, (short)0, acc, false, false);
  }

  // C/D 16x16 fp32: VGPR j -> M = j + 8*half, N = l16
  float* orow = XP + (size_t)(mt * 16 + 8 * half) * PP + pt * 16 + l16;
#pragma unroll
  for (int j = 0; j < 8; ++j) orow[(size_t)j * PP] = acc[j];
}

// ---------------------------------------------------------------------------
// Kernel 2: dt[M, D] = softplus(xp[:, 0:128] @ dt_proj_w[D, 128]^T + bias)
// fp32 WMMA, K = 128, softplus fused into the epilogue.
// ---------------------------------------------------------------------------
__device__ __forceinline__ float softplus_f(float v) {
  // numerically stable: max(v,0) + log1p(exp(-|v|))
  return fmaxf(v, 0.0f) + log1pf(__expf(-fabsf(v)));
}

__global__ __launch_bounds__(32) void ssm_dtproj_wmma(
    const float* __restrict__ XP,    // [MM, PP] (use first RR columns)
    const float* __restrict__ W2,    // [DD, RR]
    const float* __restrict__ bias,  // [DD]
    float* __restrict__ DT) {        // [MM, DD]
  const int mt   = blockIdx.x;       // 0..255
  const int dt_t = blockIdx.y;       // 0..127 (D tiles)
  const int lane = threadIdx.x;
  const int half = lane >> 4;
  const int l16  = lane & 15;

  const float* arow = XP + (size_t)(mt * 16 + l16) * PP;  // K-contiguous
  const float* brow = W2 + (size_t)(dt_t * 16 + l16) * RR;

  v8f acc = {};
  for (int k = 0; k < RR; k += 4) {
    v2f a = *(const v2f*)(arow + k + half * 2);
    v2f b;
    b.x = brow[k + half];
    b.y = brow[k + 2 + half];
    acc = __builtin_amdgcn_wmma_f32_16x16x4_f32(
        false, a, false, b, (short)0, acc, false, false);
  }

  const float bv = bias[dt_t * 16 + l16];
  float* orow = DT + (size_t)(mt * 16 + 8 * half) * DD + dt_t * 16 + l16;
#pragma unroll
  for (int j = 0; j < 8; ++j) orow[(size_t)j * DD] = softplus_f(acc[j] + bv);
}

// ---------------------------------------------------------------------------
// Kernel 3: selective scan. One thread per (b, d) chain; h[16] and a[16] in
// registers. The d-invariant B/C vectors are staged into LDS with CDNA5
// async global->LDS copies (GLOBAL_LOAD_ASYNC_TO_LDS_B128, ASYNCcnt),
// double-buffered so the next 64-step chunk streams in while the current
// one is consumed. s_wait_asynccnt 2 releases the oldest batch (in-order
// completion), acting as a split barrier ahead of __syncthreads().
// ---------------------------------------------------------------------------
#define CHUNK   64
#define NCHUNK  (LL / CHUNK)

#define S_WAIT_ASYNCCNT_0() asm volatile("s_wait_asynccnt 0x0" ::: "memory")
#define S_WAIT_ASYNCCNT_2() asm volatile("s_wait_asynccnt 0x2" ::: "memory")

// Copy CHUNK rows x 32 floats (8 KB) of B|C into an LDS buffer.
// 512 B128 transfers; 256 threads issue 2 async loads each.
__device__ __forceinline__ void async_copy_bc(
    const float* __restrict__ XP, size_t mbase, int l0,
    float* ldsbuf, int tid) {
#pragma unroll
  for (int i = 0; i < 2; ++i) {
    const int idx = tid + i * 256;   // 0..511
    const int row = idx >> 3;        // 0..63
    const int seg = idx & 7;         // 8 x 16B per 128B row
    const float* g = XP + (mbase + l0 + row) * PP + RR + seg * 4;
    // LDS aperture low 32 bits are zero -> truncated generic address is the
    // LDS byte offset the VDST VGPR of the async op expects.
    const unsigned lds_off =
        (unsigned)(uintptr_t)(ldsbuf + row * (2 * NN) + seg * 4);
    asm volatile("global_load_async_to_lds_b128 %0, %1, off"
                 :: "v"(lds_off), "v"(g) : "memory");
  }
}

__global__ __launch_bounds__(256) void ssm_scan(
    const float* __restrict__ X,      // [BB, LL, DD]
    const float* __restrict__ XP,     // [MM, PP] (cols 128..159 = B|C)
    const float* __restrict__ DT,     // [MM, DD]
    const float* __restrict__ A_log,  // [DD, NN]
    const float* __restrict__ Dp,     // [DD]
    float* __restrict__ Y) {          // [BB, LL, DD]
  const int gid = blockIdx.x * 256 + threadIdx.x;  // 0..4095
  const int b   = gid >> 11;                        // gid / DD
  const int d   = gid & (DD - 1);                   // gid % DD
  const int tid = threadIdx.x;

  __shared__ __align__(16) float sBC[2][CHUNK][2 * NN];  // 2 x 8 KB

  // a[n] = -exp(A_log[d][n])
  float a[NN];
#pragma unroll
  for (int n = 0; n < NN; ++n) a[n] = -__expf(A_log[d * NN + n]);

  float h[NN];
#pragma unroll
  for (int n = 0; n < NN; ++n) h[n] = 0.0f;

  const float dp = Dp[d];
  const size_t mbase = (size_t)b * LL;

  // Prologue: start streaming chunk 0.
  async_copy_bc(XP, mbase, 0, &sBC[0][0][0], tid);

  for (int ci = 0; ci < NCHUNK; ++ci) {
    const int buf = ci & 1;
    if (ci + 1 < NCHUNK) {
      // Kick off the next chunk into the other buffer, then wait only for
      // the *older* batch (2 ops) to land.
      async_copy_bc(XP, mbase, (ci + 1) * CHUNK, &sBC[buf ^ 1][0][0], tid);
      S_WAIT_ASYNCCNT_2();
    } else {
      S_WAIT_ASYNCCNT_0();
    }
    __syncthreads();  // all waves' chunk-ci data visible in LDS

    const int l0 = ci * CHUNK;
    for (int t = 0; t < CHUNK; ++t) {
      const size_t m  = mbase + l0 + t;
      const float dtv = DT[m * DD + d];
      const float xv  = X[m * DD + d];
      const float dtx = dtv * xv;
      float acc = 0.0f;
#pragma unroll
      for (int n = 0; n < NN; ++n) {
        const float abar = __expf(dtv * a[n]);
        h[n] = abar * h[n] + dtx * sBC[buf][t][n];
        acc += h[n] * sBC[buf][t][NN + n];
      }
      Y[m * DD + d] = acc + xv * dp;
    }
    __syncthreads();  // done reading buf before it is overwritten next iter
  }
}

// ---------------------------------------------------------------------------
extern "C" void kernel_launch(void* const* d_in, const int* in_sizes, int n_in,
                              void* d_out, int out_size, void* d_ws, size_t ws_size,
                              hipStream_t stream) {
  const float* x         = (const float*)d_in[0];  // [B,L,D]
  const float* x_proj_w  = (const float*)d_in[1];  // [160, D]
  const float* dt_proj_w = (const float*)d_in[2];  // [D, R]
  const float* dt_proj_b = (const float*)d_in[3];  // [D]
  const float* A_log     = (const float*)d_in[4];  // [D, N]
  const float* D_param   = (const float*)d_in[5];  // [D]
  float* y = (float*)d_out;

  float* xp = (float*)d_ws;                 // MM*PP floats (2.62 MB)
  float* dt = xp + (size_t)MM * PP;         // MM*DD floats (33.6 MB)

  dim3 gA(MM / 16, PP / 16);   // 256 x 10
  hipLaunchKernelGGL(ssm_xproj_wmma, gA, dim3(32), 0, stream, x, x_proj_w, xp);

  dim3 gB(MM / 16, DD / 16);   // 256 x 128
  hipLaunchKernelGGL(ssm_dtproj_wmma, gB, dim3(32), 0, stream,
                     xp, dt_proj_w, dt_proj_b, dt);

  hipLaunchKernelGGL(ssm_scan, dim3((BB * DD) / 256), dim3(256), 0, stream,
                     x, xp, dt, A_log, D_param, y);
}